// HdcGenericEncoder_18253611008356
// MI455X (gfx1250) — compile-verified
//
#include <hip/hip_runtime.h>
#include <stddef.h>

// ---------------- problem constants ----------------
#define D_DIM   10000
#define T_DIM   2048
#define L_DIM   100
#define C_DIM   4
#define KM      6          // MFCC_COV_NUM
#define FM      91         // MFCC_COV_FEAT_LENGTH
#define TN      (T_DIM - 2)  // 2046 n-gram rows

typedef __attribute__((ext_vector_type(2))) float v2f;
typedef __attribute__((ext_vector_type(8))) float v8f;

// ---------------- kernel 0: zero the sample accumulator ----------------
__global__ void zero_kernel(float* __restrict__ p, int n) {
    int i = blockIdx.x * blockDim.x + threadIdx.x;
    if (i < n) p[i] = 0.0f;
}

// ---------------- kernel 1: per_t + 3-gram rolled product + time reduction ----
// Block = 128 threads, owns D-slice [d0, d0+128) with a 2-wide halo (mod D).
// level_weight tile (100 x 130 floats = 52 KB) staged in LDS once; the whole
// time loop runs out of LDS. Time axis split 8-ways (grid.y) with an overlap
// of 2 steps; partial sums merged with global_atomic_add_f32.
#define K1_BLOCK 128
#define K1_HALO  (K1_BLOCK + 2)
#define TSPLIT   8
#define TCHUNK   256   // TSPLIT*TCHUNK >= TN

__global__ __launch_bounds__(K1_BLOCK)
void ngram_accum_kernel(const float* __restrict__ signals,
                        const float* __restrict__ keys_hv,
                        const float* __restrict__ level_weight,
                        float* __restrict__ sample_acc)
{
    __shared__ float lw[L_DIM * K1_HALO];          // level tile, stride 130
    __shared__ float cur[K1_HALO];                 // per_t for current t
    __shared__ unsigned char sidx[(TCHUNK + 2) * C_DIM];

    const int tid = threadIdx.x;
    const int d0  = blockIdx.x * K1_BLOCK;
    const int nd  = (D_DIM - d0 < K1_BLOCK) ? (D_DIM - d0) : K1_BLOCK;
    const int hd  = nd + 2;

    // ---- preload level_weight tile: columns (d0-2 .. d0+nd-1) mod D ----
    for (int j = tid; j < L_DIM * K1_HALO; j += K1_BLOCK) {
        int l    = j / K1_HALO;
        int slot = j - l * K1_HALO;
        float v = 0.0f;
        if (slot < hd) {
            int dg = d0 - 2 + slot;
            dg += (dg < 0) ? D_DIM : 0;
            dg -= (dg >= D_DIM) ? D_DIM : 0;
            v = level_weight[l * D_DIM + dg];
        }
        lw[j] = v;
    }

    // ---- per-thread channel keys for its slot(s) ----
    float km[C_DIM] = {0.f, 0.f, 0.f, 0.f};
    float ke[C_DIM] = {0.f, 0.f, 0.f, 0.f};
    if (tid < hd) {
        int dg = d0 - 2 + tid;
        dg += (dg < 0) ? D_DIM : 0;
        dg -= (dg >= D_DIM) ? D_DIM : 0;
        #pragma unroll
        for (int c = 0; c < C_DIM; ++c) km[c] = keys_hv[c * D_DIM + dg];
    }
    const int slot2 = K1_BLOCK + tid;
    if (tid < 2 && slot2 < hd) {
        int dg = d0 - 2 + slot2;
        dg -= (dg >= D_DIM) ? D_DIM : 0;
        #pragma unroll
        for (int c = 0; c < C_DIM; ++c) ke[c] = keys_hv[c * D_DIM + dg];
    }

    // ---- time chunk for this block ----
    const int ts = blockIdx.y * TCHUNK;
    const int te = (TN < ts + TCHUNK) ? TN : (ts + TCHUNK);
    const int nt = te + 2 - ts;   // #per_t steps computed here

    // ---- precompute level indices for this chunk into LDS (uint8) ----
    for (int j = tid; j < nt * C_DIM; j += K1_BLOCK) {
        int tt = ts + j / C_DIM;
        int cc = j - (j / C_DIM) * C_DIM;
        float s = signals[tt * C_DIM + cc];
        s = fminf(fmaxf(s, 0.0f), 1.0f);
        sidx[j] = (unsigned char)(int)rintf(s * (float)(L_DIM - 1));
    }
    __syncthreads();

    // ---- streaming time loop: register delay line for the rolled 3-gram ----
    float acc = 0.0f;
    float p1 = 0.0f, p2a = 0.0f, p2b = 0.0f;
    for (int t = ts; t < te + 2; ++t) {
        const int ib = (t - ts) * C_DIM;
        int i0 = sidx[ib + 0], i1 = sidx[ib + 1], i2 = sidx[ib + 2], i3 = sidx[ib + 3];
        if (tid < hd) {
            float v = km[0] * lw[i0 * K1_HALO + tid];
            v = fmaf(km[1], lw[i1 * K1_HALO + tid], v);
            v = fmaf(km[2], lw[i2 * K1_HALO + tid], v);
            v = fmaf(km[3], lw[i3 * K1_HALO + tid], v);
            cur[tid] = v;
        }
        if (tid < 2 && slot2 < hd) {
            float v = ke[0] * lw[i0 * K1_HALO + slot2];
            v = fmaf(ke[1], lw[i1 * K1_HALO + slot2], v);
            v = fmaf(ke[2], lw[i2 * K1_HALO + slot2], v);
            v = fmaf(ke[3], lw[i3 * K1_HALO + slot2], v);
            cur[slot2] = v;
        }
        __syncthreads();
        float v0 = 0.f, v1 = 0.f, v2 = 0.f;
        if (tid < nd) { v0 = cur[tid]; v1 = cur[tid + 1]; v2 = cur[tid + 2]; }
        // ngram[tg=t-2][d] = per_t[tg][d-2] * per_t[tg+1][d-1] * per_t[tg+2][d]
        if (t >= ts + 2) acc += p2b * p1 * v2;
        p2b = p2a; p2a = v0; p1 = v1;
        __syncthreads();
    }
    if (tid < nd) atomicAdd(&sample_acc[d0 + tid], acc);
}

// ---------------- kernel 2: mproj[k][d] = mfcc_w[k,d,:] . mf[k,:] via WMMA ----
// One wave per 16-row D-tile per k, chained V_WMMA_F32_16X16X4_F32.
// Layout-robust formulation: the A (16x4, M x K) layout is documented
// unambiguously (lanes 0-15: VGPR0=K0, VGPR1=K1; lanes 16-31: VGPR0=K2,
// VGPR1=K3).  The B (4x16, K x N) slot<->K mapping has two plausible variants;
// the slots that map to the SAME K under both are (VGPR0, lanes 0-15) = K0 and
// (VGPR1, lanes 16-31) = K3.  So each WMMA carries only 2 useful K values
// (K0 <- f=2i, K3 <- f=2i+1), with the mf value replicated across all N
// columns so every output column holds the result.  46 chained WMMAs cover
// K=91 (tail zero-padded); the result is read from lanes 0 (M=0..7) and
// 16 (M=8..15), VGPRs 0..7.
__global__ __launch_bounds__(256)
void mfcc_gemv_wmma_kernel(const float* __restrict__ feat,
                           const float* __restrict__ mfcc_w,
                           float* __restrict__ mproj)
{
    __shared__ float smf[FM];
    const int k = blockIdx.y;
    if (threadIdx.x < FM) smf[threadIdx.x] = feat[k * FM + threadIdx.x];
    __syncthreads();

    const int wave = threadIdx.x >> 5;
    const int lane = threadIdx.x & 31;
    const int tile = blockIdx.x * 8 + wave;
    if (tile >= (D_DIM / 16)) return;          // wave-uniform exit (EXEC stays full)

    const int d0   = tile * 16;
    const int row  = lane & 15;
    const int half = lane >> 4;
    const float* __restrict__ wp =
        mfcc_w + ((size_t)k * D_DIM + (size_t)(d0 + row)) * FM;

    v8f c = {};
    #pragma unroll
    for (int i = 0; i < 46; ++i) {             // 46 chained WMMAs, 2 K-vals each
        const int f0 = 2 * i;                  // -> K0 slot
        const int f1 = 2 * i + 1;              // -> K3 slot
        // A: only K0 (lanes 0-15, VGPR0) and K3 (lanes 16-31, VGPR1) non-zero
        v2f a; a.x = 0.0f; a.y = 0.0f;
        if (half == 0) { a.x = wp[f0]; }
        else           { a.y = (f1 < FM) ? wp[f1] : 0.0f; }
        // B: rows K0 / K3 carry mf replicated across all 16 N columns;
        // the layout-ambiguous K1/K2 slots are zero under both mappings.
        v2f b; b.x = 0.0f; b.y = 0.0f;
        if (half == 0) { b.x = smf[f0]; }
        else           { b.y = (f1 < FM) ? smf[f1] : 0.0f; }
        c = __builtin_amdgcn_wmma_f32_16x16x4_f32(
                /*neg_a=*/false, a, /*neg_b=*/false, b,
                /*c_mod=*/(short)0, c, /*reuse_a=*/false, /*reuse_b=*/false);
    }

    if (row == 0) {                            // lanes 0 (M=0..7) and 16 (M=8..15)
        float* __restrict__ out = mproj + (size_t)k * D_DIM + d0 + half * 8;
        #pragma unroll
        for (int i = 0; i < 8; ++i) out[i] = c[i];
    }
}

// ---------------- kernel 3: sinusoid feature HVs + comb + hard quantize ------
__global__ __launch_bounds__(256)
void finalize_kernel(const float* __restrict__ feat,
                     const float* __restrict__ feat_w,
                     const float* __restrict__ feat_b,
                     const float* __restrict__ mfcc_b,
                     const float* __restrict__ sample_acc,
                     const float* __restrict__ mproj,
                     float* __restrict__ out)
{
    const int d = blockIdx.x * blockDim.x + threadIdx.x;
    if (d >= D_DIM) return;

    // mfcc_hv = prod_k cos(mproj+b) * sin(mproj)
    float mh = 1.0f;
    #pragma unroll
    for (int k = 0; k < KM; ++k) {
        float p = mproj[k * D_DIM + d];
        mh *= cosf(p + mfcc_b[k * D_DIM + d]) * sinf(p);
    }

    const int cf[29] = {547, 548, 549, 551, 554, 556, 557, 558, 559, 560, 561,
                        562, 563, 565, 566, 567, 570, 576, 580, 581, 582, 583,
                        584, 585, 588, 593, 598, 599, 600};
    float h[29];
    #pragma unroll
    for (int i = 0; i < 29; ++i) {
        float s = feat[cf[i] - 1];
        float p = s * feat_w[i * D_DIM + d];
        h[i] = cosf(p + feat_b[i * D_DIM + d]) * sinf(p);
    }

    float comb = h[0] * h[8] * h[13]
               + h[1] * h[9] * h[14]
               + h[2] * h[10] * h[15]
               + h[3] * h[4]
               + h[5] * h[7] * h[22] * h[6] * h[23] * h[19] * h[18]
                      * h[20] * h[21] * h[26] * h[28] * h[27]
               + h[11] + h[12]
               + h[16] * h[24]
               + h[17] + h[25]
               + mh;

    float sh = sample_acc[d] * comb;
    out[d] = (sh > 0.0f) ? 1.0f : -1.0f;
}

// ---------------- launch ----------------
extern "C" void kernel_launch(void* const* d_in, const int* in_sizes, int n_in,
                              void* d_out, int out_size, void* d_ws, size_t ws_size,
                              hipStream_t stream) {
    (void)in_sizes; (void)n_in; (void)out_size; (void)ws_size;
    const float* signals      = (const float*)d_in[0];  // [2048,4]
    const float* feat         = (const float*)d_in[1];  // [600]
    const float* keys_hv      = (const float*)d_in[2];  // [4,10000]
    const float* level_weight = (const float*)d_in[3];  // [100,10000]
    const float* feat_w       = (const float*)d_in[4];  // [29,10000]
    const float* feat_b       = (const float*)d_in[5];  // [29,10000]
    const float* mfcc_w       = (const float*)d_in[6];  // [6,10000,91]
    const float* mfcc_b       = (const float*)d_in[7];  // [6,10000]
    float* out = (float*)d_out;

    float* ws         = (float*)d_ws;
    float* sample_acc = ws;            // D floats
    float* mproj      = ws + D_DIM;    // 6*D floats

    zero_kernel<<<(D_DIM + 255) / 256, 256, 0, stream>>>(sample_acc, D_DIM);

    dim3 g1((D_DIM + K1_BLOCK - 1) / K1_BLOCK, TSPLIT);
    ngram_accum_kernel<<<g1, K1_BLOCK, 0, stream>>>(signals, keys_hv,
                                                    level_weight, sample_acc);

    dim3 g2((D_DIM / 16 + 7) / 8, KM);
    mfcc_gemv_wmma_kernel<<<g2, 256, 0, stream>>>(feat, mfcc_w, mproj);

    finalize_kernel<<<(D_DIM + 255) / 256, 256, 0, stream>>>(
        feat, feat_w, feat_b, mfcc_b, sample_acc, mproj, out);
}